// GNN_Encoder_73212012528437
// MI455X (gfx1250) — compile-verified
//
#include <hip/hip_runtime.h>
#include <hip/hip_bf16.h>

// ---------------------------------------------------------------------------
// GNN encoder for MI455X (gfx1250, wave32).
// GEMMs use V_WMMA_F32_16X16X4_F32 (exact f32 matrix math -> no precision
// loss vs the f32 reference; workload is aggregation/bandwidth bound anyway).
// B operand is staged in LDS pre-swizzled into WMMA fragment order so each
// fragment is a single aligned ds_load_b64 into an even VGPR pair (no repack
// movs in the inner loop).  Graph aggregation uses a device-built CSR
// (count -> single-block scan -> atomic bucket fill) reused by all 3 convs.
// ---------------------------------------------------------------------------

#define NAREA  10000
#define NFEAT  16
#define HID    64
#define NEDGE  320000
#define BATCH  64
#define MROWS  (BATCH * NAREA)          // 640000 rows for every GEMM

static constexpr float EPS   = 1e-5f;
static constexpr float SLOPE = 0.01f;

typedef float v2f __attribute__((ext_vector_type(2)));
typedef float v8f __attribute__((ext_vector_type(8)));

// ---------------------------------------------------------------- CSR build

__global__ void zero_cnt_kernel(int* cnt) {
    int i = blockIdx.x * blockDim.x + threadIdx.x;
    if (i < NAREA) cnt[i] = 0;
}

__global__ void count_kernel(const int* __restrict__ ei, int* __restrict__ cnt) {
    int e = blockIdx.x * blockDim.x + threadIdx.x;
    if (e < NEDGE) atomicAdd(&cnt[ei[NEDGE + e]], 1);   // dst row of edge_index
}

// exclusive prefix sum over NAREA counts, single block of 1024 threads
__global__ void scan_kernel(const int* __restrict__ cnt,
                            int* __restrict__ ptr, int* __restrict__ cursor) {
    __shared__ int buf[1024];
    int tid = threadIdx.x;
    int carry = 0;
    for (int base = 0; base < NAREA; base += 1024) {
        int i = base + tid;
        int v = (i < NAREA) ? cnt[i] : 0;
        buf[tid] = v;
        __syncthreads();
        for (int off = 1; off < 1024; off <<= 1) {
            int t = (tid >= off) ? buf[tid - off] : 0;
            __syncthreads();
            buf[tid] += t;
            __syncthreads();
        }
        int incl  = buf[tid];
        int total = buf[1023];
        int excl  = carry + incl - v;
        if (i < NAREA) { ptr[i] = excl; cursor[i] = excl; }
        carry += total;
        __syncthreads();
    }
    if (tid == 0) ptr[NAREA] = carry;
}

__global__ void dinv_kernel(const int* __restrict__ cnt, float* __restrict__ dinv) {
    int n = blockIdx.x * blockDim.x + threadIdx.x;
    if (n < NAREA) {
        float deg = (float)(cnt[n] + 1);                // +1 self loop, >= 1
        dinv[n] = rsqrtf(deg);
    }
}

__global__ void fill_kernel(const int* __restrict__ ei,
                            int* __restrict__ cursor, int* __restrict__ csrc) {
    int e = blockIdx.x * blockDim.x + threadIdx.x;
    if (e < NEDGE) {
        int dst = ei[NEDGE + e];
        int pos = atomicAdd(&cursor[dst], 1);
        csrc[pos] = ei[e];                              // src row of edge_index
    }
}

// ------------------------------------------------------------ WMMA f32 GEMM
// C[M,64] = A[M,K] * B[K,64]  (optionally + bias, tanh for the final layer).
// Block = 256 threads = 8 waves; each wave owns a 16x64 output tile -> 128
// rows per block.  Uses V_WMMA_F32_16X16X4_F32.
//
// Fragment layout (ISA 7.12.2, 32-bit 16x4 A): lane L handles row L%16,
// holds K = {0,1} (lanes 0-15) or K = {2,3} (lanes 16-31).  B mirrors this
// over its 16 columns; C/D: VGPR r, lanes 0-15 -> row r, lanes 16-31 ->
// row r+8, col = L%16.
//
// LDS swizzle: Bs[(k>>1)*128 + col*2 + (k&1)] = B[k][col] so each lane's
// fragment (B[ka][col], B[ka+1][col]) is one aligned 8-byte ds_load_b64.

template<int K, bool TANH>
__global__ __launch_bounds__(256)
void gemm_wmma_kernel(const float* __restrict__ A, const float* __restrict__ Bm,
                      const float* __restrict__ bias, float* __restrict__ C) {
    __shared__ float Bs[K * HID];
    for (int i = threadIdx.x; i < K * HID; i += 256) {
        const int k   = i >> 6;         // row of B (K dim)
        const int col = i & 63;         // column of B (N dim)
        Bs[(k >> 1) * 128 + col * 2 + (k & 1)] = Bm[i];
    }
    __syncthreads();

    const int wave  = threadIdx.x >> 5;
    const int lane  = threadIdx.x & 31;
    const int row16 = lane & 15;
    const int hi    = lane >> 4;

    const int mbase = (blockIdx.x * 8 + wave) * 16;     // MROWS % 128 == 0
    const float* arow = A + (size_t)(mbase + row16) * K;
    const float* bfrag = &Bs[row16 * 2 + hi * 128];     // + (k0>>1)*128 per step

    v8f c[4] = {v8f{}, v8f{}, v8f{}, v8f{}};

#pragma unroll
    for (int k0 = 0; k0 < K; k0 += 4) {
        const int ka = k0 + 2 * hi;
        const v2f a = *(const v2f*)(arow + ka);                 // global_load_b64
#pragma unroll
        for (int nt = 0; nt < 4; ++nt) {
            const v2f b = *(const v2f*)(bfrag + (k0 >> 1) * 128 + nt * 32);  // ds_load_b64
            c[nt] = __builtin_amdgcn_wmma_f32_16x16x4_f32(
                false, a, false, b, (short)0, c[nt], false, false);
        }
    }

#pragma unroll
    for (int nt = 0; nt < 4; ++nt) {
        const int col = nt * 16 + row16;
        const float bcol = TANH ? bias[col] : 0.0f;
#pragma unroll
        for (int r = 0; r < 8; ++r) {
            const int rr = mbase + r + 8 * hi;
            float v = c[nt][r];
            if (TANH) v = tanhf(v + bcol);
            C[(size_t)rr * HID + col] = v;
        }
    }
}

// ------------------------------------------------- fused aggregate+epilogue
// One 64-thread block per (node n, batch b).  Gathers CSR neighbours of n
// (weight dinv[n]*dinv[src]) plus the self loop (dinv[n]^2), adds bias, then
// optional LayerNorm (reduce over H=64 in LDS), LeakyReLU, optional residual.

template<bool DO_LN, bool ADD_RES>
__global__ __launch_bounds__(HID)
void agg_kernel(const float* __restrict__ hw,
                const int* __restrict__ ptr, const int* __restrict__ csrc,
                const float* __restrict__ dinv,
                const float* __restrict__ bias,
                const float* __restrict__ gamma, const float* __restrict__ beta,
                const float* __restrict__ res, float* __restrict__ out) {
    const int n = blockIdx.x;
    const int b = blockIdx.y;
    const int h = threadIdx.x;

    const float* hwb = hw + (size_t)b * NAREA * HID;
    const float  dn  = dinv[n];

    float acc = dn * dn * hwb[(size_t)n * HID + h];     // self loop
    const int e0 = ptr[n], e1 = ptr[n + 1];
    for (int e = e0; e < e1; ++e) {
        const int s = csrc[e];
        acc += dn * dinv[s] * hwb[(size_t)s * HID + h];
    }
    acc += bias[h];

    __shared__ float red[HID];
    if (DO_LN) {
        red[h] = acc;
        __syncthreads();
#pragma unroll
        for (int off = 32; off > 0; off >>= 1) {
            if (h < off) red[h] += red[h + off];
            __syncthreads();
        }
        const float mu = red[0] * (1.0f / HID);
        __syncthreads();
        const float d = acc - mu;
        red[h] = d * d;
        __syncthreads();
#pragma unroll
        for (int off = 32; off > 0; off >>= 1) {
            if (h < off) red[h] += red[h + off];
            __syncthreads();
        }
        const float var = red[0] * (1.0f / HID);
        acc = d * rsqrtf(var + EPS) * gamma[h] + beta[h];
    }

    acc = (acc >= 0.0f) ? acc : SLOPE * acc;            // leaky_relu

    const size_t oidx = ((size_t)b * NAREA + n) * HID + h;
    if (ADD_RES) acc += res[oidx];
    out[oidx] = acc;
}

// ----------------------------------------------------------------- launcher

static inline size_t align256(size_t x) { return (x + 255) & ~(size_t)255; }

extern "C" void kernel_launch(void* const* d_in, const int* in_sizes, int n_in,
                              void* d_out, int out_size, void* d_ws, size_t ws_size,
                              hipStream_t stream) {
    const float* state = (const float*)d_in[0];
    const int*   ei    = (const int*)  d_in[1];
    // d_in[2] batch_size, d_in[3] rej_rate, d_in[4] theta_value: unused (fixed)
    const float* W1 = (const float*)d_in[5];
    const float* b1 = (const float*)d_in[6];
    const float* W2 = (const float*)d_in[7];
    const float* b2 = (const float*)d_in[8];
    const float* W3 = (const float*)d_in[9];
    const float* b3 = (const float*)d_in[10];
    const float* g1 = (const float*)d_in[11];
    const float* be1= (const float*)d_in[12];
    const float* g3 = (const float*)d_in[13];
    const float* be3= (const float*)d_in[14];
    const float* Wf = (const float*)d_in[15];
    const float* bf = (const float*)d_in[16];
    float* out = (float*)d_out;

    // workspace carve-up
    char*  ws  = (char*)d_ws;
    const size_t big = align256((size_t)MROWS * HID * sizeof(float)); // 163.84 MB
    float* buf_hw = (float*)(ws);
    float* buf_h1 = (float*)(ws + big);
    float* buf_h2 = (float*)(ws + 2 * big);
    char* p = ws + 3 * big;
    int*   cnt    = (int*)p;  p += align256(NAREA * sizeof(int));
    int*   csrp   = (int*)p;  p += align256((NAREA + 1) * sizeof(int));
    int*   cursor = (int*)p;  p += align256(NAREA * sizeof(int));
    float* dinv   = (float*)p; p += align256(NAREA * sizeof(float));
    int*   csrc   = (int*)p;  p += align256(NEDGE * sizeof(int));
    (void)ws_size; (void)in_sizes; (void)n_in; (void)out_size;

    const int TB = 256;
    const int nblkN = (NAREA + TB - 1) / TB;
    const int nblkE = (NEDGE + TB - 1) / TB;

    // 1) degree / CSR build (reused by all three convs)
    zero_cnt_kernel<<<nblkN, TB, 0, stream>>>(cnt);
    count_kernel<<<nblkE, TB, 0, stream>>>(ei, cnt);
    scan_kernel<<<1, 1024, 0, stream>>>(cnt, csrp, cursor);
    dinv_kernel<<<nblkN, TB, 0, stream>>>(cnt, dinv);
    fill_kernel<<<nblkE, TB, 0, stream>>>(ei, cursor, csrc);

    const dim3 gemm_grid(MROWS / 128);          // 8 waves * 16 rows per block
    const dim3 agg_grid(NAREA, BATCH);

    // 2) conv1: hw = x @ W1 ; h1 = leaky(LN(agg(hw) + b1))
    gemm_wmma_kernel<NFEAT, false><<<gemm_grid, 256, 0, stream>>>(state, W1, nullptr, buf_hw);
    agg_kernel<true, false><<<agg_grid, HID, 0, stream>>>(
        buf_hw, csrp, csrc, dinv, b1, g1, be1, nullptr, buf_h1);

    // 3) conv2: hw = h1 @ W2 ; h2 = leaky(agg(hw) + b2) + h1
    gemm_wmma_kernel<HID, false><<<gemm_grid, 256, 0, stream>>>(buf_h1, W2, nullptr, buf_hw);
    agg_kernel<false, true><<<agg_grid, HID, 0, stream>>>(
        buf_hw, csrp, csrc, dinv, b2, nullptr, nullptr, buf_h1, buf_h2);

    // 4) conv3: hw = h2 @ W3 ; h3 = leaky(LN(agg(hw) + b3)) + h1   (h3 -> buf_h2)
    gemm_wmma_kernel<HID, false><<<gemm_grid, 256, 0, stream>>>(buf_h2, W3, nullptr, buf_hw);
    agg_kernel<true, true><<<agg_grid, HID, 0, stream>>>(
        buf_hw, csrp, csrc, dinv, b3, g3, be3, buf_h1, buf_h2);

    // 5) out = tanh(h3 @ Wf + bf)
    gemm_wmma_kernel<HID, true><<<gemm_grid, 256, 0, stream>>>(buf_h2, Wf, bf, out);
}